// MaskedSlidingPSN_35107062678295
// MI455X (gfx1250) — compile-verified
//
#include <hip/hip_runtime.h>

typedef float v8f __attribute__((ext_vector_type(8)));
typedef float v2f __attribute__((ext_vector_type(2)));

#define ORDER  32
#define TDIM   1024
#define DDIM   1024
#define BDIM   16
#define TGROUP 8    // consecutive 16-row t-tiles per wave (rolling k-window)
#define NCHUNK 12   // K chunks of 4 covering window k in [t0-32, t0+15]

// One wave owns a [TGROUP*16 (t) x 16 (d)] strip of the output for one batch b.
// h[b,t,d] = sum_{0<=delta<32} weight[31-delta] * x[b,t-delta,d] + threshold
// spike   = (h >= 0) ? 1 : 0
__global__ __launch_bounds__(256) void psn_wmma_f32_kernel(
    const float* __restrict__ x,     // [B, T, D]
    const float* __restrict__ w,     // [32]
    const float* __restrict__ thr,   // [1]
    float* __restrict__ out)         // [B, T, D]
{
    const int lane = threadIdx.x & 31;
    // Force wave-uniform values into SGPRs so tile bases are scalar and all
    // global accesses become saddr + one voffset VGPR + immediate offset.
    const int wave = __builtin_amdgcn_readfirstlane(threadIdx.x >> 5);
    const int tile = blockIdx.x * 8 + wave;

    const int nDD = DDIM / 16;               // 64
    const int nTG = TDIM / (16 * TGROUP);    // 8
    const int dd  = tile % nDD;              // innermost: adjacent waves -> adjacent d
    const int tg  = (tile / nDD) % nTG;
    const int b   = tile / (nDD * nTG);

    const int d0 = dd * 16;
    const int g0 = tg * (16 * TGROUP);

    const int n  = lane & 15;   // A row (M) / B,D column (N)
    const int hi = lane >> 4;   // K-pair select within a chunk

    // ---- Toeplitz A chunks: position independent, built once. ----
    // A layout (16x4 f32): VGPR0 = K(2*hi), VGPR1 = K(2*hi+1), rows M = lane&15.
    v2f a[NCHUNK];
#pragma unroll
    for (int c = 0; c < NCHUNK; ++c) {
        const int krel  = 4 * c - 32 + 2 * hi;   // k - t0 of the .x element
        const int diff0 = n - krel;              // t - k
        const int diff1 = diff0 - 1;
        int i0 = ORDER - 1 - diff0; i0 = i0 < 0 ? 0 : (i0 > ORDER - 1 ? ORDER - 1 : i0);
        int i1 = ORDER - 1 - diff1; i1 = i1 < 0 ? 0 : (i1 > ORDER - 1 ? ORDER - 1 : i1);
        const float w0 = w[i0];                  // unconditional clamped load
        const float w1 = w[i1];
        a[c].x = (diff0 >= 0 && diff0 < ORDER) ? w0 : 0.0f;
        a[c].y = (diff1 >= 0 && diff1 < ORDER) ? w1 : 0.0f;
    }

    const float threshold = thr[0];

    // Scalar tile bases (element (t=g0, d=d0) of batch b).
    const float* xbase = x   + (b * TDIM + g0) * DDIM + d0;
    float*       obase = out + (b * TDIM + g0) * DDIM + d0;
    // Per-lane element offsets; row contribution is a compile-time immediate.
    const int voff = hi * 2 * DDIM + n;  // B loads: .x element of a chunk
    const int soff = hi * 8 * DDIM + n;  // D stores

    // ---- Preload the first tile's 12 B chunks (k in [g0-32, g0+15]). ----
    // Logical chunk c at step s lives in physical slot (4*s + c) % 12.
    v2f bb[NCHUNK];
    if (g0 == 0) {
        // Scalar branch: only the first t-group has k<0 rows (read as zero).
#pragma unroll
        for (int c = 0; c < NCHUNK; ++c) {
            const int rel0 = 4 * c - 32 + 2 * hi;     // == k since g0 == 0
            const int rel1 = rel0 + 1;
            const float v0 = xbase[(rel0 < 0 ? 0 : rel0) * DDIM + n];
            const float v1 = xbase[(rel1 < 0 ? 0 : rel1) * DDIM + n];
            bb[c].x = (rel0 < 0) ? 0.0f : v0;
            bb[c].y = (rel1 < 0) ? 0.0f : v1;
        }
    } else {
#pragma unroll
        for (int c = 0; c < NCHUNK; ++c) {
            bb[c].x = xbase[(4 * c - 32) * DDIM + voff];
            bb[c].y = xbase[(4 * c - 32 + 1) * DDIM + voff];
        }
    }

#pragma unroll
    for (int s = 0; s < TGROUP; ++s) {
        // Prefetch next subtile's 4 new chunks (rows g0+16(s+1) .. +15) into
        // temporaries; issued before the WMMAs so loads overlap compute.
        v2f nb[4];
        if (s < TGROUP - 1) {
#pragma unroll
            for (int j = 0; j < 4; ++j) {
                nb[j].x = xbase[(16 * (s + 1) + 4 * j) * DDIM + voff];
                nb[j].y = xbase[(16 * (s + 1) + 4 * j + 1) * DDIM + voff];
            }
        }

        // Two independent WMMA chains; threshold folded into acc0 init.
        v8f acc0 = {threshold, threshold, threshold, threshold,
                    threshold, threshold, threshold, threshold};
        v8f acc1 = {};
#pragma unroll
        for (int c = 0; c < NCHUNK; c += 2) {
            acc0 = __builtin_amdgcn_wmma_f32_16x16x4_f32(
                false, a[c],     false, bb[(4 * s + c)     % NCHUNK],
                (short)0, acc0, false, false);
            acc1 = __builtin_amdgcn_wmma_f32_16x16x4_f32(
                false, a[c + 1], false, bb[(4 * s + c + 1) % NCHUNK],
                (short)0, acc1, false, false);
        }

        // Heaviside epilogue + store. D layout: col N = lane&15, row = v + 8*hi.
#pragma unroll
        for (int v = 0; v < 8; ++v) {
            const float h = acc0[v] + acc1[v];
            obase[(16 * s + v) * DDIM + soff] = (h >= 0.0f) ? 1.0f : 0.0f;
        }

        // Retire logical chunks 0..3 of this step: their physical slots
        // ((4s+j)%12) are exactly next step's logical chunks 8..11.
        if (s < TGROUP - 1) {
#pragma unroll
            for (int j = 0; j < 4; ++j) bb[(4 * s + j) % NCHUNK] = nb[j];
        }
    }
}

extern "C" void kernel_launch(void* const* d_in, const int* in_sizes, int n_in,
                              void* d_out, int out_size, void* d_ws, size_t ws_size,
                              hipStream_t stream) {
    (void)in_sizes; (void)n_in; (void)out_size; (void)d_ws; (void)ws_size;
    const float* x   = (const float*)d_in[0];   // [16,1024,1024] f32
    const float* w   = (const float*)d_in[1];   // [32] f32
    const float* thr = (const float*)d_in[2];   // [1] f32
    float* out = (float*)d_out;

    // 16 b * 8 t-groups * 64 d-tiles = 8192 waves; 8 waves / 256-thread block.
    const int total_waves = BDIM * (TDIM / (16 * TGROUP)) * (DDIM / 16);
    const int blocks = total_waves / 8;
    psn_wmma_f32_kernel<<<blocks, 256, 0, stream>>>(x, w, thr, out);
}